// SimpleGNN_2Layer_1760936591465
// MI455X (gfx1250) — compile-verified
//
#include <hip/hip_runtime.h>

typedef __attribute__((ext_vector_type(2))) float v2f;
typedef __attribute__((ext_vector_type(8))) float v8f;

// ---------------------------------------------------------------------------
// Degree / normalization
// ---------------------------------------------------------------------------
__global__ void zero_i32(int* __restrict__ p, int n) {
  int i = blockIdx.x * blockDim.x + threadIdx.x;
  if (i < n) p[i] = 0;
}

__global__ void deg_count(const long long* __restrict__ dst, int* __restrict__ deg, int E) {
  int i = blockIdx.x * blockDim.x + threadIdx.x;
  if (i < E) atomicAdd(&deg[(int)dst[i]], 1);
}

__global__ void compute_dinv(const int* __restrict__ deg, float* __restrict__ dinv, int n) {
  int i = blockIdx.x * blockDim.x + threadIdx.x;
  if (i < n) dinv[i] = rsqrtf((float)deg[i] + 1.0f);  // +1 = self loop; always > 0
}

// ---------------------------------------------------------------------------
// Dense projection Y[M,N] = X[M,K] @ W[K,N] using V_WMMA_F32_16X16X4_F32.
// One wave computes one 16x16 output tile; K stepped by 4.
// A fragment (16x4 f32): lanes 0-15 hold {K0,K1} of row M0+lane,
//                        lanes 16-31 hold {K2,K3} of row M0+(lane-16).
// B fragment (4x16 f32): lanes 0-15 hold {B[k0][n],B[k1][n]}, lanes 16-31
//                        hold {B[k2][n],B[k3][n]}, n = lane%16.
// C/D (16x16 f32, 8 VGPRs): VGPR r -> row r (lanes 0-15) / row r+8 (16-31).
// ---------------------------------------------------------------------------
__global__ void gemm_wmma_f32(const float* __restrict__ X, const float* __restrict__ W,
                              float* __restrict__ Y, int M, int K, int N) {
  const int wavesPerBlock = blockDim.x >> 5;
  const int waveId = blockIdx.x * wavesPerBlock + (threadIdx.x >> 5);
  const int lane = threadIdx.x & 31;
  const int ntiles = N >> 4;
  const int mtiles = (M + 15) >> 4;
  const int mt = waveId / ntiles;
  const int nt = waveId % ntiles;
  if (mt >= mtiles) return;                    // wave-uniform exit: EXEC stays full
  const int row0 = mt << 4;
  const int col0 = nt << 4;
  const int hl = lane & 15;
  const bool hi = lane >= 16;
  int arow = row0 + hl;
  if (arow >= M) arow = M - 1;                 // clamp loads; stores guarded below

  v8f c = {};
  for (int k = 0; k < K; k += 4) {
    const int ka = k + (hi ? 2 : 0);
    v2f a, b;
    a.x = X[(size_t)arow * K + ka];
    a.y = X[(size_t)arow * K + ka + 1];
    b.x = W[(size_t)ka * N + col0 + hl];
    b.y = W[(size_t)(ka + 1) * N + col0 + hl];
    c = __builtin_amdgcn_wmma_f32_16x16x4_f32(
        /*neg_a=*/false, a, /*neg_b=*/false, b,
        /*c_mod=*/(short)0, c, /*reuse_a=*/false, /*reuse_b=*/false);
  }

  const int rbase = row0 + (hi ? 8 : 0);
#pragma unroll
  for (int r = 0; r < 8; ++r) {
    const int row = rbase + r;
    if (row < M) Y[(size_t)row * N + col0 + hl] = c[r];
  }
}

// ---------------------------------------------------------------------------
// agg[v,f] = h[v,f] * dinv[v]^2   (self-loop message; also initializes agg)
// ---------------------------------------------------------------------------
__global__ void selfloop_init(const float* __restrict__ h, const float* __restrict__ dinv,
                              float* __restrict__ agg, int n, int lgF) {
  int i = blockIdx.x * blockDim.x + threadIdx.x;
  if (i >= (n << lgF)) return;
  const int v = i >> lgF;
  float w = dinv[v];
  agg[i] = h[i] * (w * w);
}

// ---------------------------------------------------------------------------
// agg[dst] += h[src] * dinv[src]*dinv[dst]; (1<<lgChunks) threads per edge,
// each handling 4 features (float4 gather + 4 f32 atomics into L2).
// ---------------------------------------------------------------------------
__global__ void edge_scatter(const long long* __restrict__ src, const long long* __restrict__ dstv,
                             const float* __restrict__ dinv, const float* __restrict__ h,
                             float* __restrict__ agg, int E, int F, int lgChunks) {
  long long idx = (long long)blockIdx.x * blockDim.x + threadIdx.x;
  long long e = idx >> lgChunks;
  if (e >= E) return;
  const int cf = ((int)idx & ((1 << lgChunks) - 1)) << 2;
  const int s = (int)src[e];
  const int d = (int)dstv[e];
  const float w = dinv[s] * dinv[d];
  const float4 hv = *reinterpret_cast<const float4*>(h + (size_t)s * F + cf);
  float* out = agg + (size_t)d * F + cf;
  atomicAdd(out + 0, hv.x * w);
  atomicAdd(out + 1, hv.y * w);
  atomicAdd(out + 2, hv.z * w);
  atomicAdd(out + 3, hv.w * w);
}

__global__ void bias_relu(float* __restrict__ agg, const float* __restrict__ b, int n, int F) {
  int i = blockIdx.x * blockDim.x + threadIdx.x;
  if (i >= n * F) return;
  float v = agg[i] + b[i & (F - 1)];
  agg[i] = v > 0.0f ? v : 0.0f;
}

// ---------------------------------------------------------------------------
// out[v] = dot(h[v,0:16], Wl) + bl
// ---------------------------------------------------------------------------
__global__ void final_linear(const float* __restrict__ h, const float* __restrict__ Wl,
                             const float* __restrict__ bl, float* __restrict__ out, int n) {
  int v = blockIdx.x * blockDim.x + threadIdx.x;
  if (v >= n) return;
  const float4* hp = reinterpret_cast<const float4*>(h + (size_t)v * 16);
  const float4* wp = reinterpret_cast<const float4*>(Wl);
  float acc = bl[0];
#pragma unroll
  for (int j = 0; j < 4; ++j) {
    const float4 hv = hp[j];
    const float4 wv = wp[j];
    acc += hv.x * wv.x + hv.y * wv.y + hv.z * wv.z + hv.w * wv.w;
  }
  out[v] = acc;
}

// ---------------------------------------------------------------------------
extern "C" void kernel_launch(void* const* d_in, const int* in_sizes, int n_in,
                              void* d_out, int out_size, void* d_ws, size_t ws_size,
                              hipStream_t stream) {
  const float*     x   = (const float*)d_in[0];
  const long long* ei  = (const long long*)d_in[1];   // int64 edge_index [2,E] flat
  const float*     W1  = (const float*)d_in[2];
  const float*     b1  = (const float*)d_in[3];
  const float*     W2  = (const float*)d_in[4];
  const float*     b2  = (const float*)d_in[5];
  const float*     Wl  = (const float*)d_in[6];
  const float*     bl  = (const float*)d_in[7];
  float*           out = (float*)d_out;

  const int N = in_sizes[0] / 128;   // 100000 nodes
  const int E = in_sizes[1] / 2;     // 3200000 edges
  const long long* src = ei;         // edge_index[0]
  const long long* dst = ei + E;     // edge_index[1]

  // Bump-allocate scratch (≈40 MB total) from d_ws, 256B aligned.
  char* ws = (char*)d_ws;
  size_t off = 0;
  auto bump = [&](size_t bytes) -> void* {
    void* p = ws + off;
    off += (bytes + 255) & ~(size_t)255;
    return p;
  };
  int*   deg  = (int*)  bump((size_t)N * 4);
  float* dinv = (float*)bump((size_t)N * 4);
  float* h1   = (float*)bump((size_t)N * 32 * 4);
  float* agg1 = (float*)bump((size_t)N * 32 * 4);
  float* h2   = (float*)bump((size_t)N * 16 * 4);
  float* agg2 = (float*)bump((size_t)N * 16 * 4);
  (void)ws_size; (void)n_in; (void)out_size;

  const int T = 256;

  // Degree / normalization
  zero_i32<<<(N + T - 1) / T, T, 0, stream>>>(deg, N);
  deg_count<<<(E + T - 1) / T, T, 0, stream>>>(dst, deg, E);
  compute_dinv<<<(N + T - 1) / T, T, 0, stream>>>(deg, dinv, N);

  // Layer 1: h1 = X @ W1  (M=N, K=128, Nout=32)
  {
    const int tiles = ((N + 15) / 16) * (32 / 16);
    gemm_wmma_f32<<<(tiles + 3) / 4, 128, 0, stream>>>(x, W1, h1, N, 128, 32);
  }
  selfloop_init<<<((N * 32) + T - 1) / T, T, 0, stream>>>(h1, dinv, agg1, N, 5);
  {
    const long long tot = (long long)E * 8;   // 8 chunk-threads per edge (F=32)
    edge_scatter<<<(int)((tot + T - 1) / T), T, 0, stream>>>(src, dst, dinv, h1, agg1, E, 32, 3);
  }
  bias_relu<<<((N * 32) + T - 1) / T, T, 0, stream>>>(agg1, b1, N, 32);

  // Layer 2: h2 = relu1 @ W2  (M=N, K=32, Nout=16)
  {
    const int tiles = ((N + 15) / 16) * (16 / 16);
    gemm_wmma_f32<<<(tiles + 3) / 4, 128, 0, stream>>>(agg1, W2, h2, N, 32, 16);
  }
  selfloop_init<<<((N * 16) + T - 1) / T, T, 0, stream>>>(h2, dinv, agg2, N, 4);
  {
    const long long tot = (long long)E * 4;   // 4 chunk-threads per edge (F=16)
    edge_scatter<<<(int)((tot + T - 1) / T), T, 0, stream>>>(src, dst, dinv, h2, agg2, E, 16, 2);
  }
  bias_relu<<<((N * 16) + T - 1) / T, T, 0, stream>>>(agg2, b2, N, 16);

  // Readout: out = h2r @ Wl + bl
  final_linear<<<(N + T - 1) / T, T, 0, stream>>>(agg2, Wl, bl, out, N);
}